// afm_27805618274715
// MI455X (gfx1250) — compile-verified
//
#include <hip/hip_runtime.h>
#include <math.h>

typedef __attribute__((ext_vector_type(2))) float v2f;
typedef __attribute__((ext_vector_type(8))) float v8f;

#define GPB     64            // groups per block
#define ROWS    (GPB * 3)     // 192 pair-product rows
#define VSTRIDE 68            // padded row stride in dwords (68 % 64 = 4 -> no bank conflicts)
#define CSTR    212           // carr column stride (212 % 64 = 20 -> conflict-free b128 stores)
#define BLOCK   128           // 4 waves; 3 tiles per wave

// sigmoid via hardware v_exp_f32 + v_rcp_f32 (avoids IEEE div expansion)
__device__ __forceinline__ float fast_sigmoid(float x) {
    return __builtin_amdgcn_rcpf(1.0f + __expf(-x));
}

__global__ __launch_bounds__(BLOCK) void afm_kernel(
    const int*   __restrict__ group,
    const float* __restrict__ embed,
    const float* __restrict__ W1,
    const float* __restrict__ b1,
    const float* __restrict__ W2,
    const float* __restrict__ b2,
    const float* __restrict__ Wp,
    const float* __restrict__ bp,
    float*       __restrict__ out)
{
    __shared__ __align__(16) float vbuf[ROWS * VSTRIDE]; // 52224 B
    __shared__ __align__(16) float carr[4 * CSTR];       // C cols 0..3, col-major
    __shared__ float w1s[256];

    const int tid = threadIdx.x;
    const long long gbase = (long long)blockIdx.x * GPB;

    // ---- stage W1 (64x4 row-major) into LDS ----
    w1s[tid]       = W1[tid];
    w1s[tid + 128] = W1[tid + 128];

    // ---- gather + pairwise products: 2 threads per group, 32 dims each ----
    {
        const int gs = tid >> 1;       // group slot 0..63
        const int h  = tid & 1;        // dim half
        const long long g = gbase + gs;
        const int i0 = group[g * 3 + 0];
        const int i1 = group[g * 3 + 1];
        const int i2 = group[g * 3 + 2];
        const float4* e0 = (const float4*)(embed + (long long)i0 * 64 + h * 32);
        const float4* e1 = (const float4*)(embed + (long long)i1 * 64 + h * 32);
        const float4* e2 = (const float4*)(embed + (long long)i2 * 64 + h * 32);
        const int r0  = gs * 3;
        const int off = h * 32;
        #pragma unroll
        for (int c = 0; c < 8; ++c) {
            float4 a = e0[c], b = e1[c], d = e2[c];
            float4 p01 = make_float4(a.x * b.x, a.y * b.y, a.z * b.z, a.w * b.w);
            float4 p02 = make_float4(a.x * d.x, a.y * d.y, a.z * d.z, a.w * d.w);
            float4 p12 = make_float4(b.x * d.x, b.y * d.y, b.z * d.z, b.w * d.w);
            *(float4*)&vbuf[(r0 + 0) * VSTRIDE + off + 4 * c] = p01;
            *(float4*)&vbuf[(r0 + 1) * VSTRIDE + off + 4 * c] = p02;
            *(float4*)&vbuf[(r0 + 2) * VSTRIDE + off + 4 * c] = p12;
        }
    }
    __syncthreads();

    // ---- WMMA phase: A = v rows (16x64 tiles), B = W1 padded to 16 cols ----
    const int lane = tid & 31;
    const int wave = tid >> 5;     // 0..3
    const int n    = lane & 15;    // row index (A) / col index (B, C)
    const int kh   = lane >> 4;    // K half: lanes 16-31 carry K+2 / K+3 and rows M+8

    // Prebuild B operands for all 16 k-steps:
    //   VGPR0: K=4kk+2*kh, VGPR1: K=4kk+1+2*kh, col = n (zero for n>=4)
    v2f breg[16];
    #pragma unroll
    for (int kk = 0; kk < 16; ++kk) {
        float bx = 0.0f, by = 0.0f;
        if (n < 4) {
            bx = w1s[(4 * kk + 0 + 2 * kh) * 4 + n];
            by = w1s[(4 * kk + 1 + 2 * kh) * 4 + n];
        }
        breg[kk][0] = bx;
        breg[kk][1] = by;
    }

    #pragma unroll
    for (int j = 0; j < 3; ++j) {
        const int tile  = wave * 3 + j;
        const int rbase = tile * 16;
        const float* arow = &vbuf[(rbase + n) * VSTRIDE + 2 * kh];

        v8f c0 = {};
        v8f c1 = {};
        #pragma unroll
        for (int kk = 0; kk < 8; ++kk) {
            v2f a = *(const v2f*)(arow + 4 * kk);
            c0 = __builtin_amdgcn_wmma_f32_16x16x4_f32(
                false, a, false, breg[kk], (short)0, c0, false, false);
        }
        #pragma unroll
        for (int kk = 8; kk < 16; ++kk) {
            v2f a = *(const v2f*)(arow + 4 * kk);
            c1 = __builtin_amdgcn_wmma_f32_16x16x4_f32(
                false, a, false, breg[kk], (short)0, c1, false, false);
        }

        // Dump C columns 0..3 to LDS (col-major, padded stride), one exec
        // toggle + two b128 stores per tile. Row id = rbase + kh*8 + r.
        float cs0 = c0[0] + c1[0];
        float cs1 = c0[1] + c1[1];
        float cs2 = c0[2] + c1[2];
        float cs3 = c0[3] + c1[3];
        float cs4 = c0[4] + c1[4];
        float cs5 = c0[5] + c1[5];
        float cs6 = c0[6] + c1[6];
        float cs7 = c0[7] + c1[7];
        if (n < 4) {
            float* p = &carr[n * CSTR + rbase + kh * 8];
            *(float4*)(p + 0) = make_float4(cs0, cs1, cs2, cs3);
            *(float4*)(p + 4) = make_float4(cs4, cs5, cs6, cs7);
        }
    }
    __syncthreads();

    // ---- per-group epilogue: sigmoid MLP + Linear(3,1) + sigmoid*2-1 ----
    if (tid < GPB) {
        const float b10 = b1[0], b11 = b1[1], b12 = b1[2], b13 = b1[3];
        const float w20 = W2[0], w21 = W2[1], w22 = W2[2], w23 = W2[3];
        const float b2v = b2[0];
        float acc[3];
        #pragma unroll
        for (int p = 0; p < 3; ++p) {
            const int row = tid * 3 + p;
            float s0 = fast_sigmoid(carr[0 * CSTR + row] + b10);
            float s1 = fast_sigmoid(carr[1 * CSTR + row] + b11);
            float s2 = fast_sigmoid(carr[2 * CSTR + row] + b12);
            float s3 = fast_sigmoid(carr[3 * CSTR + row] + b13);
            acc[p] = b2v + s0 * w20 + s1 * w21 + s2 * w22 + s3 * w23;
        }
        float logit = acc[0] * Wp[0] + acc[1] * Wp[1] + acc[2] * Wp[2] + bp[0];
        out[gbase + tid] = fast_sigmoid(logit) * 2.0f - 1.0f;
    }
}

extern "C" void kernel_launch(void* const* d_in, const int* in_sizes, int n_in,
                              void* d_out, int out_size, void* d_ws, size_t ws_size,
                              hipStream_t stream) {
    const int*   group = (const int*)  d_in[0];
    const float* embed = (const float*)d_in[1];
    const float* W1    = (const float*)d_in[2];
    const float* b1    = (const float*)d_in[3];
    const float* W2    = (const float*)d_in[4];
    const float* b2    = (const float*)d_in[5];
    const float* Wp    = (const float*)d_in[6];
    const float* bp    = (const float*)d_in[7];
    float* out = (float*)d_out;

    const int ngroups = in_sizes[0] / 3;   // 262144
    const int blocks  = ngroups / GPB;     // 4096

    afm_kernel<<<blocks, BLOCK, 0, stream>>>(group, embed, W1, b1, W2, b2, Wp, bp, out);
}